// Model_70025146794076
// MI455X (gfx1250) — compile-verified
//
#include <hip/hip_runtime.h>

typedef float v2f __attribute__((ext_vector_type(2)));
typedef float v8f __attribute__((ext_vector_type(8)));

#define NELEM 65536          // 256*256
#define FGRID 26250          // 5*25*7*6*5

// ---------------------------------------------------------------------------
// K0: zero the 8-float accumulator block in workspace
// ---------------------------------------------------------------------------
__global__ void zero_ws(float* ws) {
    if (threadIdx.x < 8) ws[threadIdx.x] = 0.0f;
}

// ---------------------------------------------------------------------------
// K1: s[n] = sum_k x[n,k]*theta[k] (n=0..4) and abssum = sum_k |theta[k]|
// via V_WMMA_F32_16X16X4_F32.
//   A (16x4):  rows 0..4 = x chunks, row 5 = sign(theta) chunk, rows 6..15 = 0
//   B (4x16):  every column = theta chunk (broadcast)
//   D[m,0] accumulates; lane 0 ends with D[0..5,0] in c[0..5].
// Branchless inner loop: every lane loads a *valid* address; the A fragment
// is selected with cndmasks (no EXEC-mask divergence around the WMMA).
// 256 waves, each covering a contiguous K-span of 256 (64 WMMAs).
// ---------------------------------------------------------------------------
__device__ __forceinline__ float sgnf(float t) {
    // branchless sign: +-1 with t's sign, 0 if t == 0
    return (t != 0.0f) ? __builtin_copysignf(1.0f, t) : 0.0f;
}

__global__ __launch_bounds__(256) void dot_wmma(const float* __restrict__ x,
                                                const float* __restrict__ th,
                                                float* __restrict__ ws) {
    const int lane   = threadIdx.x & 31;
    const int waveId = (int)((blockIdx.x * blockDim.x + threadIdx.x) >> 5);
    const int m  = lane & 15;              // A-matrix row
    const int k0 = ((lane >> 4) & 1) << 1; // K sub-offset: 0 (lanes 0-15), 2 (lanes 16-31)

    const bool isX    = (m < 5);
    const bool isSign = (m == 5);
    // Safe per-lane A source: lanes m>=5 read x row 4 (valid address, masked out).
    const float* aptr = x + (isX ? m : 4) * NELEM;

    int kb = waveId * 256;                 // this wave's K base
    v8f c = {};

    #pragma unroll 4
    for (int t = 0; t < 64; ++t, kb += 4) {
        const int k = kb + k0;
        const v2f av = *(const v2f*)(aptr + k);   // global_load_b64
        const v2f bv = *(const v2f*)(th + k);     // global_load_b64 (B fragment)
        // A fragment: x value, sign(theta), or 0 — pure cndmask selection.
        v2f A;
        A.x = isX ? av.x : (isSign ? sgnf(bv.x) : 0.0f);
        A.y = isX ? av.y : (isSign ? sgnf(bv.y) : 0.0f);
        c = __builtin_amdgcn_wmma_f32_16x16x4_f32(
                /*neg_a=*/false, A, /*neg_b=*/false, bv,
                /*c_mod=*/(short)0, c, /*reuse_a=*/false, /*reuse_b=*/false);
    }

    if (lane == 0) {
        #pragma unroll
        for (int i = 0; i < 6; ++i) atomicAdd(&ws[i], c[i]); // ws[0..4]=s[n], ws[5]=abssum
    }
}

// ---------------------------------------------------------------------------
// K2: bulk copy (the bandwidth-dominant part, ~210 MB):
//   out[1 .. 1+65536)              = theta
//   out[1+65536 .. 1+65536+gridN)  = grid (row j overwritten later by K3)
// 16B-aligned b128 loads; dword stores (dest is only 4B-aligned at +65537).
// ---------------------------------------------------------------------------
__global__ __launch_bounds__(256) void copy_out(const float* __restrict__ th,
                                                const float* __restrict__ grid,
                                                float* __restrict__ out,
                                                long grid4) {
    const long t   = (long)blockIdx.x * blockDim.x + threadIdx.x;
    const long th4 = NELEM / 4;
    if (t < th4) {
        const float4 v = ((const float4*)th)[t];
        long o = 1 + t * 4;
        out[o] = v.x; out[o + 1] = v.y; out[o + 2] = v.z; out[o + 3] = v.w;
    } else if (t < th4 + grid4) {
        const long g = t - th4;
        const float4 v = ((const float4*)grid)[g];
        long o = 1 + NELEM + g * 4;
        out[o] = v.x; out[o + 1] = v.y; out[o + 2] = v.z; out[o + 3] = v.w;
    }
}

// ---------------------------------------------------------------------------
// K3: f_grid + loss. Single block of 256 threads handles 26,250 points,
// writes f_grid into output grid row j, block-reduces the weighted sum,
// and writes out[0] = sum(f*W4) - 0.5*abssum.
// ---------------------------------------------------------------------------
__global__ __launch_bounds__(256) void fgrid_loss(const float* __restrict__ ws,
                                                  const int* __restrict__ jptr,
                                                  float* __restrict__ out) {
    __shared__ float red[256];
    const int j = jptr[0];
    float* gout = out + 1 + NELEM + (long)j * FGRID;

    const float dx = 0.26179938779914946f;  // 2*pi/24
    const float dy = 0.03f;                 // 0.18/6
    const float dz = 0.036f;                // 0.18/5
    const float dw = 0.05f;                 // 0.2/4

    float part = 0.0f;
    for (int idx = threadIdx.x; idx < FGRID; idx += 256) {
        int d = idx % 5;
        int r = idx / 5;
        int cc = r % 6; r /= 6;
        int b  = r % 7; r /= 7;
        int a  = r % 25; r /= 25;
        int n  = r;                          // 0..4

        float s  = ws[n];
        float xa = dx * (float)a;
        float yb = -0.09f + dy * (float)b;
        float zc = -0.09f + dz * (float)cc;
        float wd = 0.9f + dw * (float)d;

        float arg = s * cosf(xa) * wd + yb + zc;
        float f   = expf(-arg * arg);
        gout[idx] = f;

        float wgt = dx * ((a == 0 || a == 24) ? 0.5f : 1.0f)
                  * dy * ((b == 0 || b == 6)  ? 0.5f : 1.0f)
                  * dz * ((cc == 0 || cc == 5)? 0.5f : 1.0f)
                  * dw * ((d == 0 || d == 4)  ? 0.5f : 1.0f);
        part += f * wgt;
    }

    red[threadIdx.x] = part;
    __syncthreads();
    for (int s = 128; s > 0; s >>= 1) {
        if ((int)threadIdx.x < s) red[threadIdx.x] += red[threadIdx.x + s];
        __syncthreads();
    }
    if (threadIdx.x == 0) out[0] = red[0] - 0.5f * ws[5];
}

// ---------------------------------------------------------------------------
extern "C" void kernel_launch(void* const* d_in, const int* in_sizes, int n_in,
                              void* d_out, int out_size, void* d_ws, size_t ws_size,
                              hipStream_t stream) {
    const float* x    = (const float*)d_in[0];   // [5,256,256]
    const float* th   = (const float*)d_in[1];   // [1,256,256]
    const float* grid = (const float*)d_in[2];   // [2000,5,25,7,6,5]
    const int*   jp   = (const int*)d_in[3];     // scalar
    float* out = (float*)d_out;
    float* ws  = (float*)d_ws;

    const long gridN = (long)in_sizes[2];
    const long grid4 = gridN / 4;                // 52,500,000 divisible by 4
    const long th4   = NELEM / 4;
    const long tot   = th4 + grid4;
    const int  cblocks = (int)((tot + 255) / 256);

    zero_ws   <<<1, 32, 0, stream>>>(ws);
    dot_wmma  <<<32, 256, 0, stream>>>(x, th, ws);
    copy_out  <<<cblocks, 256, 0, stream>>>(th, grid, out, grid4);
    fgrid_loss<<<1, 256, 0, stream>>>(ws, jp, out);
}